// RecurrentGCN_50946902065580
// MI455X (gfx1250) — compile-verified
//
#include <hip/hip_runtime.h>
#include <hip/hip_bf16.h>
#include <math.h>

#define N_NODES 131072
#define N_EDGES 4194304
#define F_NODE  16
#define F_EDGE  8
#define HID     32

typedef __attribute__((ext_vector_type(2))) float v2f;
typedef __attribute__((ext_vector_type(8))) float v8f;

__device__ __forceinline__ float sigmoidf_(float v) {
    return 1.0f / (1.0f + __expf(-v));
}

// ---------------------------------------------------------------------------
// Zero a float region (deg + dinv + tx1). Grid-stride so it is launch-size
// independent and graph-capture safe.
// ---------------------------------------------------------------------------
__global__ __launch_bounds__(256) void zero_f32(float* __restrict__ p, int n) {
    int i = blockIdx.x * blockDim.x + threadIdx.x;
    int stride = gridDim.x * blockDim.x;
    for (; i < n; i += stride) p[i] = 0.0f;
}

// ---------------------------------------------------------------------------
// Edge-weight MLP: ew = sigmoid(relu(ea @ W1 + b1) @ W2 + b2)
// One wave32 handles 16 edges. Layer 1 via V_WMMA_F32_16X16X4_F32:
//   A = 16 edges x 4 feats (2 K-chunks cover F_EDGE=8)
//   B = W1 chunk 4 x 16    (2 column halves cover 32 hidden)
// Layer 2 (32 -> 1) via per-lane FMA + 16-lane butterfly reduction.
// ---------------------------------------------------------------------------
__global__ __launch_bounds__(256) void edge_mlp(
        const float* __restrict__ ea, const float* __restrict__ w1,
        const float* __restrict__ b1, const float* __restrict__ w2,
        const float* __restrict__ b2, float* __restrict__ ew) {
    const int lane = threadIdx.x & 31;
    const int half = lane >> 4;   // 0: K pair {0,1}/rows 0-7, 1: K pair {2,3}/rows 8-15
    const int l16  = lane & 15;
    const int wid  = (blockIdx.x * blockDim.x + threadIdx.x) >> 5;
    const int eb   = wid * 16;    // wave-uniform
    if (eb >= N_EDGES) return;

    // A fragments: row M = eb + l16, lane<16 holds K = k0+{0,1}, lane>=16 K = k0+{2,3}
    const float* rowp = ea + (size_t)(eb + l16) * F_EDGE + 2 * half;
    v2f a0 = *(const v2f*)(rowp + 0);  // K-chunk 0 (K 0..3)
    v2f a1 = *(const v2f*)(rowp + 4);  // K-chunk 1 (K 4..7)

    // B fragments: B[K][N] = w1[K*32 + N]; lane holds K = k0 + 2*half (+1), N = n0 + l16
    const int kk = 2 * half;
    v2f b00, b01, b10, b11;
    b00.x = w1[(kk + 0) * 32 + l16];      b00.y = w1[(kk + 1) * 32 + l16];
    b01.x = w1[(kk + 0) * 32 + 16 + l16]; b01.y = w1[(kk + 1) * 32 + 16 + l16];
    b10.x = w1[(kk + 4) * 32 + l16];      b10.y = w1[(kk + 5) * 32 + l16];
    b11.x = w1[(kk + 4) * 32 + 16 + l16]; b11.y = w1[(kk + 5) * 32 + 16 + l16];

    v8f c0 = {}; v8f c1 = {};
    c0 = __builtin_amdgcn_wmma_f32_16x16x4_f32(false, a0, false, b00, (short)0, c0, false, false);
    c0 = __builtin_amdgcn_wmma_f32_16x16x4_f32(false, a1, false, b10, (short)0, c0, false, false);
    c1 = __builtin_amdgcn_wmma_f32_16x16x4_f32(false, a0, false, b01, (short)0, c1, false, false);
    c1 = __builtin_amdgcn_wmma_f32_16x16x4_f32(false, a1, false, b11, (short)0, c1, false, false);

    // Layer 2: per lane partials, columns n = l16 (c0) and 16 + l16 (c1)
    const float b1lo = b1[l16], b1hi = b1[16 + l16];
    const float w2lo = w2[l16], w2hi = w2[16 + l16];
    float p[8];
#pragma unroll
    for (int j = 0; j < 8; ++j) {
        float h0 = fmaxf(c0[j] + b1lo, 0.0f);
        float h1 = fmaxf(c1[j] + b1hi, 0.0f);
        p[j] = h0 * w2lo + h1 * w2hi;   // row m = j + 8*half
    }
    // Butterfly sum within each 16-lane half (xor masks < 16 never cross halves)
#pragma unroll
    for (int off = 1; off < 16; off <<= 1) {
#pragma unroll
        for (int j = 0; j < 8; ++j) p[j] += __shfl_xor(p[j], off, 32);
    }
    // Select p[l16] without dynamic register indexing, lanes 0..7 of each half store
    float s = p[0];
#pragma unroll
    for (int j = 1; j < 8; ++j) s = (l16 == j) ? p[j] : s;
    if (l16 < 8) {
        int e = eb + 8 * half + l16;
        ew[e] = sigmoidf_(s + b2[0]);
    }
}

// ---------------------------------------------------------------------------
// deg[src[e]] += ew[e]
// ---------------------------------------------------------------------------
__global__ __launch_bounds__(256) void deg_accum(
        const long long* __restrict__ ei, const float* __restrict__ ew,
        float* __restrict__ deg) {
    int e = blockIdx.x * blockDim.x + threadIdx.x;
    if (e >= N_EDGES) return;
    int s = (int)ei[e];
    atomicAdd(&deg[s], ew[e]);
}

// ---------------------------------------------------------------------------
// dinv = deg > 0 ? rsqrt(deg) : 0
// ---------------------------------------------------------------------------
__global__ __launch_bounds__(256) void make_dinv(
        const float* __restrict__ deg, float* __restrict__ dinv) {
    int i = blockIdx.x * blockDim.x + threadIdx.x;
    if (i >= N_NODES) return;
    float d = deg[i];
    dinv[i] = (d > 0.0f) ? rsqrtf(d) : 0.0f;
}

// ---------------------------------------------------------------------------
// tx1[dst] += (-dinv[src]*ew*dinv[dst]) * x[src]   (16 lanes per edge, one
// lane per feature -> 64B coalesced gathers of x, f32 atomics land in L2;
// the 8MB tx1 buffer fits entirely in the 192MB L2).
// ---------------------------------------------------------------------------
__global__ __launch_bounds__(256) void scatter_tx1(
        const long long* __restrict__ ei, const float* __restrict__ ew,
        const float* __restrict__ dinv, const float* __restrict__ x,
        float* __restrict__ tx1) {
    int gid = blockIdx.x * blockDim.x + threadIdx.x;
    int e = gid >> 4;
    int f = gid & 15;
    if (e >= N_EDGES) return;
    int s = (int)ei[e];               // 16 lanes hit same address -> 1 request
    int d = (int)ei[N_EDGES + e];
    float w = -dinv[s] * ew[e] * dinv[d];
    atomicAdd(&tx1[(size_t)d * F_NODE + f], w * x[(size_t)s * F_NODE + f]);
}

// ---------------------------------------------------------------------------
// Node update: per wave, 16 nodes.
//   pre_z = [x | tx1] @ wz (32x32) ; pre_h = [x | tx1] @ wh (32x32)
//   out   = (1 - sigmoid(pre_z + bz)) * tanh(pre_h + bh)
// The [2,16,32] weight stacks are already the concatenated 32x32 B matrix.
// 32 V_WMMA_F32_16X16X4_F32 per wave; A fragments reused across both gates.
// ---------------------------------------------------------------------------
__global__ __launch_bounds__(256) void node_update(
        const float* __restrict__ x, const float* __restrict__ tx1,
        const float* __restrict__ wz, const float* __restrict__ wh,
        const float* __restrict__ bxz, const float* __restrict__ bhz,
        const float* __restrict__ bxh, const float* __restrict__ bhh,
        float* __restrict__ out) {
    const int lane = threadIdx.x & 31;
    const int half = lane >> 4;
    const int l16  = lane & 15;
    const int wid  = (blockIdx.x * blockDim.x + threadIdx.x) >> 5;
    const int nb   = wid * 16;   // wave-uniform
    if (nb >= N_NODES) return;

    // A fragments: K 0..15 from x row, K 16..31 from tx1 row
    const float* xr = x   + (size_t)(nb + l16) * F_NODE;
    const float* tr = tx1 + (size_t)(nb + l16) * F_NODE;
    v2f a[8];
#pragma unroll
    for (int kc = 0; kc < 4; ++kc) a[kc]     = *(const v2f*)(xr + 4 * kc + 2 * half);
#pragma unroll
    for (int kc = 0; kc < 4; ++kc) a[4 + kc] = *(const v2f*)(tr + 4 * kc + 2 * half);

    v8f cz0 = {}, cz1 = {}, ch0 = {}, ch1 = {};
#pragma unroll
    for (int kc = 0; kc < 8; ++kc) {
        const int k = 4 * kc + 2 * half;
        v2f bz0, bz1, bh0, bh1;
        bz0.x = wz[(k + 0) * 32 + l16];      bz0.y = wz[(k + 1) * 32 + l16];
        bz1.x = wz[(k + 0) * 32 + 16 + l16]; bz1.y = wz[(k + 1) * 32 + 16 + l16];
        bh0.x = wh[(k + 0) * 32 + l16];      bh0.y = wh[(k + 1) * 32 + l16];
        bh1.x = wh[(k + 0) * 32 + 16 + l16]; bh1.y = wh[(k + 1) * 32 + 16 + l16];
        cz0 = __builtin_amdgcn_wmma_f32_16x16x4_f32(false, a[kc], false, bz0, (short)0, cz0, false, false);
        cz1 = __builtin_amdgcn_wmma_f32_16x16x4_f32(false, a[kc], false, bz1, (short)0, cz1, false, false);
        ch0 = __builtin_amdgcn_wmma_f32_16x16x4_f32(false, a[kc], false, bh0, (short)0, ch0, false, false);
        ch1 = __builtin_amdgcn_wmma_f32_16x16x4_f32(false, a[kc], false, bh1, (short)0, ch1, false, false);
    }

    const float bz_lo = bxz[l16]      + bhz[l16];
    const float bz_hi = bxz[16 + l16] + bhz[16 + l16];
    const float bh_lo = bxh[l16]      + bhh[l16];
    const float bh_hi = bxh[16 + l16] + bhh[16 + l16];

#pragma unroll
    for (int j = 0; j < 8; ++j) {
        const int node = nb + j + 8 * half;          // D layout: M = j + 8*(lane/16)
        float z0 = sigmoidf_(cz0[j] + bz_lo);
        float z1 = sigmoidf_(cz1[j] + bz_hi);
        float t0 = tanhf(ch0[j] + bh_lo);
        float t1 = tanhf(ch1[j] + bh_hi);
        out[(size_t)node * HID + l16]      = (1.0f - z0) * t0;
        out[(size_t)node * HID + 16 + l16] = (1.0f - z1) * t1;
    }
}

// ---------------------------------------------------------------------------
extern "C" void kernel_launch(void* const* d_in, const int* in_sizes, int n_in,
                              void* d_out, int out_size, void* d_ws, size_t ws_size,
                              hipStream_t stream) {
    const float*     x   = (const float*)d_in[0];
    const long long* ei  = (const long long*)d_in[1];   // int64 [2, N_EDGES]
    const float*     ea  = (const float*)d_in[2];
    const float*     w1  = (const float*)d_in[3];
    const float*     b1  = (const float*)d_in[4];
    const float*     w2  = (const float*)d_in[5];
    const float*     b2  = (const float*)d_in[6];
    const float*     wxz = (const float*)d_in[7];
    const float*     bxz = (const float*)d_in[8];
    const float*     bhz = (const float*)d_in[10];
    const float*     wxh = (const float*)d_in[15];
    const float*     bxh = (const float*)d_in[16];
    const float*     bhh = (const float*)d_in[18];
    float* out = (float*)d_out;

    // Workspace layout (floats): ew | deg | dinv | tx1
    float* ew   = (float*)d_ws;
    float* deg  = ew + N_EDGES;
    float* dinv = deg + N_NODES;
    float* tx1  = dinv + N_NODES;

    // Zero deg + dinv + tx1 (dinv gets fully overwritten; zeroing is harmless)
    zero_f32<<<2048, 256, 0, stream>>>(deg, N_NODES * 18);

    // 1) edge MLP: 16 edges/wave, 8 waves/block -> 128 edges/block
    edge_mlp<<<N_EDGES / 128, 256, 0, stream>>>(ea, w1, b1, w2, b2, ew);

    // 2) degree accumulation
    deg_accum<<<N_EDGES / 256, 256, 0, stream>>>(ei, ew, deg);

    // 3) dinv
    make_dinv<<<N_NODES / 256, 256, 0, stream>>>(deg, dinv);

    // 4) scatter tx1 = -sum lhat_w * x[src] into dst (16 lanes per edge)
    scatter_tx1<<<(N_EDGES * 16) / 256, 256, 0, stream>>>(ei, ew, dinv, x, tx1);

    // 5) node update (gates + blend), 16 nodes/wave -> 128 nodes/block
    node_update<<<N_NODES / 128, 256, 0, stream>>>(x, tx1, wxz, wxh,
                                                   bxz, bhz, bxh, bhh, out);
}